// BlockwiseButterflyRotation_68994354643577
// MI455X (gfx1250) — compile-verified
//
#include <hip/hip_runtime.h>
#include <hip/hip_bf16.h>
#include <math.h>

// ---------------------------------------------------------------------------
// Blockwise butterfly rotation as batched GEMM on the CDNA5 f32 WMMA pipe.
// Per 256-wide block nb, the 8 butterfly stages compose into a 256x256
// orthogonal matrix M_nb;  Y = X * M_nb^T per block.
//   Kernel 1: build M_nb^T in d_ws in an LDS-ready pair-interleaved layout.
//   Kernel 2: stage B_nb (256 KB) into LDS via the Tensor Data Mover, then
//             each wave computes a 16x256 output strip with 16 chained
//             V_WMMA_F32_16X16X4_F32 accumulators, A loaded once per k-step.
// Traffic: X read once (256 MB, NT), Y written once (256 MB, NT), B staging
// ~256 MB of L2-resident traffic -> HBM-bound at ~22 us (23.3 TB/s).
// ---------------------------------------------------------------------------

#define DIMX    4096
#define BLOCKW  256
#define NBLK    16
#define NSTAGE  8
#define LDSROW  257              // padded LDS row stride for kernel 1
#define WAVES   8                // waves per workgroup in GEMM kernel
#define STRIPS  2                // 16-row strips per wave
#define TILES_PER_WG (WAVES * STRIPS)

typedef __attribute__((ext_vector_type(2))) float        v2f;
typedef __attribute__((ext_vector_type(8))) float        v8f;
typedef __attribute__((ext_vector_type(4))) unsigned int v4u;
typedef __attribute__((ext_vector_type(8))) int          v8i;
typedef __attribute__((ext_vector_type(4))) int          v4i;

// ---------------------------------------------------------------------------
// Kernel 1: build composite rotation matrix for block nb; store transposed &
// pair-interleaved:  ws[nb][(k>>1)*512 + n*2 + (k&1)] = M_nb[n][k]
// (so the GEMM's B fragment {B[k][n], B[k+1][n]} is one contiguous b64).
// grid = NBLK x 256 threads, dynamic LDS = LDSROW*256*4 bytes (~263 KB).
// ---------------------------------------------------------------------------
__global__ void butterfly_build_mats(const float* __restrict__ angles,
                                     float* __restrict__ ws) {
    extern __shared__ float M[];            // [256][LDSROW]
    const int nb  = blockIdx.x;
    const int tid = threadIdx.x;            // 0..255

    for (int j = 0; j < BLOCKW; ++j)
        M[tid * LDSROW + j] = (tid == j) ? 1.0f : 0.0f;
    __syncthreads();

    for (int s = 0; s < NSTAGE; ++s) {
        const int stride = 1 << s;
        for (int w = 0; w < BLOCKW / 2; ++w) {      // pair index == angle index
            const int g = w >> s;
            const int i = w & (stride - 1);
            const int p = g * 2 * stride + i;
            const int q = p + stride;
            const float ang = angles[(nb * NSTAGE + s) * (BLOCKW / 2) + w];
            const float c  = cosf(ang);
            const float sn = sinf(ang);
            const float a = M[p * LDSROW + tid];
            const float b = M[q * LDSROW + tid];
            M[p * LDSROW + tid] = c  * a - sn * b;
            M[q * LDSROW + tid] = sn * a + c  * b;
        }
        __syncthreads();
    }

    float* wnb = ws + (size_t)nb * BLOCKW * BLOCKW;
    for (int k = 0; k < BLOCKW; ++k)
        wnb[(k >> 1) * (2 * BLOCKW) + tid * 2 + (k & 1)] = M[tid * LDSROW + k];
}

// ---------------------------------------------------------------------------
// Kernel 2: grid = (rowTiles/16, NBLK), 256 threads (8 waves), 256 KB LDS.
// Wave layouts per CDNA5 ISA:
//   A 16x4 f32 : lanes 0-15 -> M=lane, {v0,v1}={K0,K1}; lanes 16-31 {K2,K3}
//   B 4x16 f32 : lanes 0-15 {v0,v1}={K0,K1} rows, lanes 16-31 {K2,K3}, N=lane&15
//   C 16x16 f32: VGPR v -> M = v + 8*(lane>>4), N = lane&15
// ---------------------------------------------------------------------------
__global__ void __launch_bounds__(256)
butterfly_wmma_gemm(const float* __restrict__ X,
                    const float* __restrict__ Bmats,
                    float* __restrict__ Y) {
    extern __shared__ float Blds[];          // 256 KB: [k/2][n][2] pair layout
    const int lane   = threadIdx.x & 31;
    const int waveId = threadIdx.x >> 5;
    const int nb     = blockIdx.y;

    // ---- Stage B_nb into LDS with the Tensor Data Mover (async DMA) ----
    if (threadIdx.x == 0) {
        const unsigned long long gaddr =
            (unsigned long long)(const void*)(Bmats + (size_t)nb * BLOCKW * BLOCKW);
        const unsigned int laddr = (unsigned int)(unsigned long long)(void*)Blds;

        // D# group 0: count=1 | lds_addr | global_addr[56:0] | type=2
        v4u g0;
        g0.x = 0x1u;
        g0.y = laddr;
        g0.z = (unsigned int)(gaddr & 0xFFFFFFFFull);
        g0.w = (unsigned int)((gaddr >> 32) & 0x01FFFFFFull) | (2u << 30);

        // D# group 1: data_size=4B; tensor 256x256 f32, stride 256; tile 256x256
        v8i g1;
        g1[0] = (int)(2u << 16);                  // workgroup_mask=0, data_size=2(4B)
        g1[1] = (int)((unsigned)BLOCKW << 16);    // tensor_dim0[15:0]=256
        g1[2] = (int)((unsigned)BLOCKW << 16);    // dim0 hi=0 | tensor_dim1[15:0]=256
        g1[3] = (int)((unsigned)BLOCKW << 16);    // dim1 hi=0 | tile_dim0=256
        g1[4] = BLOCKW;                           // tile_dim1=256 | tile_dim2=0
        g1[5] = BLOCKW;                           // tensor_dim0_stride=256
        g1[6] = 0;
        g1[7] = 0;

        const v4i gz4 = {0, 0, 0, 0};             // groups 2/3 unused (2D tile)
        const v8i gz8 = {0, 0, 0, 0, 0, 0, 0, 0}; // extra group (clang-23 form)
        __builtin_amdgcn_tensor_load_to_lds(g0, g1, gz4, gz4, gz8, 0);
        __builtin_amdgcn_s_wait_tensorcnt(0);
    }
    __syncthreads();

    const int m    = lane & 15;                   // A row / B-N / C col index
    const int kOff = (lane >> 4) << 1;            // 0 or 2
    const int rOff = (lane >> 4) << 3;            // 0 or 8

    for (int s = 0; s < STRIPS; ++s) {
        const int rowTile = blockIdx.x * TILES_PER_WG + s * WAVES + waveId;
        const int rowBase = rowTile * 16;
        const float* xrow = X + (size_t)(rowBase + m) * DIMX + nb * BLOCKW;

        v8f acc[NBLK];
        #pragma unroll
        for (int nt = 0; nt < NBLK; ++nt)
            acc[nt] = (v8f){0.f, 0.f, 0.f, 0.f, 0.f, 0.f, 0.f, 0.f};

        for (int k = 0; k < BLOCKW; k += 4) {
            // A fragment: streamed from HBM once, non-temporal.
            const v2f a =
                __builtin_nontemporal_load((const v2f*)(xrow + k + kOff));
            const float* bp = Blds + ((k + kOff) >> 1) * (2 * BLOCKW);
            #pragma unroll
            for (int nt = 0; nt < NBLK; ++nt) {
                const v2f b = *(const v2f*)(bp + (nt * 16 + m) * 2); // ds b64
                acc[nt] = __builtin_amdgcn_wmma_f32_16x16x4_f32(
                    /*neg_a=*/false, a, /*neg_b=*/false, b,
                    /*c_mod=*/(short)0, acc[nt],
                    /*reuse_a=*/false, /*reuse_b=*/false);
            }
        }

        float* yb = Y + (size_t)(rowBase + rOff) * DIMX + nb * BLOCKW + m;
        #pragma unroll
        for (int nt = 0; nt < NBLK; ++nt) {
            #pragma unroll
            for (int v = 0; v < 8; ++v)
                __builtin_nontemporal_store(acc[nt][v],
                                            yb + (size_t)v * DIMX + nt * 16);
        }
    }
}

extern "C" void kernel_launch(void* const* d_in, const int* in_sizes, int n_in,
                              void* d_out, int out_size, void* d_ws, size_t ws_size,
                              hipStream_t stream) {
    const float* x      = (const float*)d_in[0];
    const float* angles = (const float*)d_in[1];
    float*       out    = (float*)d_out;
    float*       mats   = (float*)d_ws;          // NBLK * 256 * 256 f32 = 4 MB

    const int rows     = in_sizes[0] / DIMX;     // 16384
    const int rowTiles = rows / 16;              // 1024

    butterfly_build_mats<<<NBLK, BLOCKW, LDSROW * BLOCKW * sizeof(float), stream>>>(
        angles, mats);

    dim3 grid(rowTiles / TILES_PER_WG, NBLK);    // (64, 16)
    butterfly_wmma_gemm<<<grid, 256, BLOCKW * BLOCKW * sizeof(float), stream>>>(
        x, mats, out);
}